// Network_21552145891481
// MI455X (gfx1250) — compile-verified
//
#include <hip/hip_runtime.h>
#include <hip/hip_bf16.h>

typedef float v2f __attribute__((ext_vector_type(2)));
typedef float v8f __attribute__((ext_vector_type(8)));

#define TILE   16
#define HALO   5
#define INW    26                 // 16 + 2*5 valid columns/rows
#define LDSW   36                 // padded row stride (2nd col-group reads up to col lo+32 <= 36)
#define CH_STRIDE (INW * LDSW)    // 936 floats per channel plane
#define NEG_SLOPE 0.2f

// Fused: 5x (residual conv3x3 + leaky) [WMMA f32 16x16x4] -> residual conv1x1+leaky
// -> outer residual -> cubic attention -> store. One wave32 per 16x16 tile.
__global__ __launch_bounds__(32) void fused_net(const float* __restrict__ x,
                                                const float* __restrict__ feature,
                                                float* __restrict__ out)
{
    const int tx   = blockIdx.x;
    const int ty   = blockIdx.y;
    const int b    = blockIdx.z;
    const int ox   = tx << 4;
    const int oy   = ty << 4;
    const int lane = threadIdx.x;
    const int o16  = lane & 15;          // M-row (A) / N-col (B) / pixel id
    const int kh2  = (lane >> 4) << 1;   // K offset for high half-wave: 0 or 2

    __shared__ float bufA[3 * CH_STRIDE];
    __shared__ float bufB[3 * CH_STRIDE];
    __shared__ float orig[3 * 256];      // central 16x16x3 for the outer residual
    __shared__ float feat[492];          // per-sample feature vector

    // ---- stage feature vector in LDS ----
    const float* fsrc = feature + (size_t)b * 492;
    for (int i = lane; i < 492; i += 32) feat[i] = fsrc[i];

    // ---- load halo-5 input tile ----
    const float* xb = x + ((size_t)b * 3 << 20);
    const bool interior = (ox >= HALO) && (oy >= HALO) &&
                          (ox + TILE + HALO <= 1024) && (oy + TILE + HALO <= 1024);
    if (interior) {
        // no bounds checks: straight global_load_b32 stream
        for (int idx = lane; idx < 3 * INW * INW; idx += 32) {
            int ch  = idx / (INW * INW);
            int rem = idx - ch * (INW * INW);
            int ry  = rem / INW;
            int rx  = rem - ry * INW;
            bufA[ch * CH_STRIDE + ry * LDSW + rx] =
                xb[((size_t)ch << 20) + ((size_t)(oy + ry - HALO) << 10) + (ox + rx - HALO)];
        }
        // zero the pad columns 26..35
        for (int idx = lane; idx < 3 * INW * (LDSW - INW); idx += 32) {
            int ch  = idx / (INW * (LDSW - INW));
            int rem = idx - ch * (INW * (LDSW - INW));
            int ry  = rem / (LDSW - INW);
            int rx  = INW + (rem - ry * (LDSW - INW));
            bufA[ch * CH_STRIDE + ry * LDSW + rx] = 0.0f;
        }
    } else {
        for (int idx = lane; idx < 3 * CH_STRIDE; idx += 32) {
            int ch  = idx / CH_STRIDE;
            int rem = idx - ch * CH_STRIDE;
            int ry  = rem / LDSW;
            int rx  = rem - ry * LDSW;
            float v = 0.0f;
            if (rx < INW) {
                int gy = oy + ry - HALO;
                int gx = ox + rx - HALO;
                if (gy >= 0 && gy < 1024 && gx >= 0 && gx < 1024)
                    v = xb[((size_t)ch << 20) + ((size_t)gy << 10) + gx];
            }
            bufA[idx] = v;
        }
    }
    __syncthreads();   // single wave -> S_NOP; LDS is in-order per wave

    // ---- stash original central tile for the outer residual ----
    for (int i = lane; i < 768; i += 32) {
        int ch = i >> 8, rem = i & 255, rr = rem >> 4, cc = rem & 15;
        orig[i] = bufA[ch * CH_STRIDE + (rr + HALO) * LDSW + (cc + HALO)];
    }

    // ---- precompute per-lane im2col B offsets (k -> ci,dy,dx), pad k>=27 -> k=26;
    //      the matching A column is zero so the padded B row contributes nothing ----
    int offB0[7], offB1[7];
#pragma unroll
    for (int kb = 0; kb < 7; ++kb) {
        int k0 = kb * 4 + kh2;
        int k1 = k0 + 1;
        int kk0 = (k0 < 27) ? k0 : 26;
        int kk1 = (k1 < 27) ? k1 : 26;
        int c0i = kk0 / 9, rm0 = kk0 - c0i * 9, dy0 = rm0 / 3, dx0 = rm0 - dy0 * 3;
        int c1i = kk1 / 9, rm1 = kk1 - c1i * 9, dy1 = rm1 / 3, dx1 = rm1 - dy1 * 3;
        offB0[kb] = c0i * CH_STRIDE + dy0 * LDSW + dx0;
        offB1[kb] = c1i * CH_STRIDE + dy1 * LDSW + dx1;
    }
    const bool arow = (o16 < 3);
    const int  arowc = arow ? o16 : 0;

    float* pin  = bufA;
    float* pout = bufB;

    // ---- 5x residual conv3x3 blocks via V_WMMA_F32_16X16X4_F32 ----
    // D[16x16] = A[16x28] x B[28x16]; M rows 0..2 = out channels,
    // K = (ci*9 + dy*3 + dx) (27 real + 1 zero pad), N = 16 output columns.
#pragma unroll
    for (int l = 0; l < 5; ++l) {
        const int lo = l + 1;            // valid output region [lo, hi)
        const int hi = 25 - l;
        const int wb = l * 84;           // layer weight base in feat[]
        const bool have2 = (lo + 16) < hi;   // compile-time after unroll

        // A operand: per-sample weights, hoisted out of the pixel loops (branchless)
        v2f Areg[7];
#pragma unroll
        for (int kb = 0; kb < 7; ++kb) {
            int k0 = kb * 4 + kh2;
            int k1 = k0 + 1;
            float a0 = feat[wb + arowc * 27 + ((k0 < 27) ? k0 : 0)];
            float a1 = feat[wb + arowc * 27 + ((k1 < 27) ? k1 : 0)];
            a0 = (arow && k0 < 27) ? a0 : 0.0f;
            a1 = (arow && k1 < 27) ? a1 : 0.0f;
            Areg[kb][0] = a0;
            Areg[kb][1] = a1;
        }
        const float bias0 = feat[wb + 81];
        const float bias1 = feat[wb + 82];
        const float bias2 = feat[wb + 83];

        for (int r = lo; r < hi; ++r) {
            // base of the im2col window for this row / group 0, per lane
            const int base = (r - 1) * LDSW + (lo - 1) + o16;
            v8f acc0 = {};
            v8f acc1 = {};
#pragma unroll
            for (int kb = 0; kb < 7; ++kb) {
                float b00 = pin[base + offB0[kb]];
                float b01 = pin[base + offB1[kb]];
                v2f Bv0 = {b00, b01};
                acc0 = __builtin_amdgcn_wmma_f32_16x16x4_f32(
                    false, Areg[kb], false, Bv0, (short)0, acc0, false, false);
                if (have2) {
                    float b10 = pin[base + 16 + offB0[kb]];
                    float b11 = pin[base + 16 + offB1[kb]];
                    v2f Bv1 = {b10, b11};
                    acc1 = __builtin_amdgcn_wmma_f32_16x16x4_f32(
                        false, Areg[kb], false, Bv1, (short)0, acc1, false, false);
                }
            }
            // D layout: lane t<16 holds acc[o] = conv(out ch o, pixel lo+t)
            if (lane < 16) {
                const int col = lo + o16;           // group 0 is always full width
                {
                    float t0 = pin[0 * CH_STRIDE + r * LDSW + col] + acc0[0] + bias0;
                    float t1 = pin[1 * CH_STRIDE + r * LDSW + col] + acc0[1] + bias1;
                    float t2 = pin[2 * CH_STRIDE + r * LDSW + col] + acc0[2] + bias2;
                    pout[0 * CH_STRIDE + r * LDSW + col] = (t0 >= 0.f) ? t0 : NEG_SLOPE * t0;
                    pout[1 * CH_STRIDE + r * LDSW + col] = (t1 >= 0.f) ? t1 : NEG_SLOPE * t1;
                    pout[2 * CH_STRIDE + r * LDSW + col] = (t2 >= 0.f) ? t2 : NEG_SLOPE * t2;
                }
                if (have2) {
                    const int col1 = lo + 16 + o16;
                    if (col1 < hi) {
                        float t0 = pin[0 * CH_STRIDE + r * LDSW + col1] + acc1[0] + bias0;
                        float t1 = pin[1 * CH_STRIDE + r * LDSW + col1] + acc1[1] + bias1;
                        float t2 = pin[2 * CH_STRIDE + r * LDSW + col1] + acc1[2] + bias2;
                        pout[0 * CH_STRIDE + r * LDSW + col1] = (t0 >= 0.f) ? t0 : NEG_SLOPE * t0;
                        pout[1 * CH_STRIDE + r * LDSW + col1] = (t1 >= 0.f) ? t1 : NEG_SLOPE * t1;
                        pout[2 * CH_STRIDE + r * LDSW + col1] = (t2 >= 0.f) ? t2 : NEG_SLOPE * t2;
                    }
                }
            }
        }
        float* tmp = pin; pin = pout; pout = tmp;
    }

    // ---- conv1x1 block + outer residual + attention + store ----
    for (int p = lane; p < 256; p += 32) {
        int rr = p >> 4, cc = p & 15;
        int R = rr + HALO, C = cc + HALO;
        float x0 = pin[0 * CH_STRIDE + R * LDSW + C];
        float x1 = pin[1 * CH_STRIDE + R * LDSW + C];
        float x2 = pin[2 * CH_STRIDE + R * LDSW + C];
        float xin[3] = {x0, x1, x2};
        float v[3];
#pragma unroll
        for (int o = 0; o < 3; ++o) {
            float s = feat[420 + o * 3 + 0] * x0
                    + feat[420 + o * 3 + 1] * x1
                    + feat[420 + o * 3 + 2] * x2
                    + feat[429 + o];
            float t = xin[o] + s;
            float y = (t >= 0.0f) ? t : NEG_SLOPE * t;
            v[o] = orig[o * 256 + p] + y;   // outer residual
        }
        int gy = oy + rr, gx = ox + cc;
        float h  = (float)gy * (1.0f / 1024.0f);
        float w  = (float)gx * (1.0f / 1024.0f);
        float h2 = h * h, h3 = h2 * h, w2 = w * w, w3 = w2 * w;
#pragma unroll
        for (int c = 0; c < 3; ++c) {
            const float* f = &feat[432 + 20 * c];
            float xi = v[c];
            float xp2 = xi * xi, xp3 = xp2 * xi;
            float att = f[0] * h3      + f[1] * h2 * w  + f[2] * h2 * xi + f[3] * h2
                      + f[4] * h * w2  + f[5] * h * w * xi + f[6] * h * w
                      + f[7] * h * xp2 + f[8] * h * xi  + f[9] * h
                      + f[10] * w3     + f[11] * w2 * xi + f[12] * w2
                      + f[13] * w * xp2+ f[14] * w * xi + f[15] * w
                      + f[16] * xp3    + f[17] * xp2    + f[18] * xi + f[19];
            out[(((size_t)(b * 3 + c)) << 20) + ((size_t)gy << 10) + gx] =
                xi * (1.0f + att);
        }
    }
}

// thumb_x passthrough: vectorized float4 copy into second output segment
__global__ __launch_bounds__(256) void copy_thumb(const float4* __restrict__ src,
                                                  float4* __restrict__ dst, int n4)
{
    int i = blockIdx.x * blockDim.x + threadIdx.x;
    if (i < n4) dst[i] = src[i];
}

extern "C" void kernel_launch(void* const* d_in, const int* in_sizes, int n_in,
                              void* d_out, int out_size, void* d_ws, size_t ws_size,
                              hipStream_t stream)
{
    const float* x       = (const float*)d_in[0];   // (8,3,1024,1024)
    const float* thumb   = (const float*)d_in[1];   // (8,3,256,256)
    const float* feature = (const float*)d_in[2];   // (8,492)
    float* out = (float*)d_out;

    dim3 grid(1024 / TILE, 1024 / TILE, 8);         // 64 x 64 x 8 tiles
    fused_net<<<grid, 32, 0, stream>>>(x, feature, out);

    const int xElems = 8 * 3 * 1024 * 1024;         // 25165824
    const int tElems = 8 * 3 * 256 * 256;           // 1572864
    const int n4 = tElems / 4;
    copy_thumb<<<(n4 + 255) / 256, 256, 0, stream>>>(
        (const float4*)thumb, (float4*)out + (xElems / 4), n4);
}